// TestModel_38233798869345
// MI455X (gfx1250) — compile-verified
//
#include <hip/hip_runtime.h>
#include <hip/hip_bf16.h>
#include <stdint.h>

// CDNA5 ext-vector types for the TDM descriptor groups (amdgpu-toolchain 6-arg form:
//   __builtin_amdgcn_tensor_load_to_lds(u32x4 g0, i32x8 g1, i32x4 g2, i32x4 g3, i32x8 g4, i32 cpol))
typedef unsigned int u32x4 __attribute__((ext_vector_type(4)));
typedef int          i32x4 __attribute__((ext_vector_type(4)));
typedef int          i32x8 __attribute__((ext_vector_type(8)));

#define IN_H   64
#define IN_W   64
#define RS     18           // resized H = W
#define OUTHW  16           // deform-conv output H = W
#define KK     9            // 3x3 kernel taps
#define CC     3            // input channels
#define IMG_ELEMS (CC * IN_H * IN_W)   // 12288 floats = 48 KB

__global__ __launch_bounds__(256)
void deform_resize_kernel(const float* __restrict__ x,
                          const float* __restrict__ offset,
                          const float* __restrict__ mask,
                          const float* __restrict__ weight,
                          float* __restrict__ out)
{
    __shared__ float lds_x[IMG_ELEMS];          // raw 3x64x64 image (TDM target), 48 KB
    __shared__ float lds_r[CC * RS * RS];       // resized 3x18x18 image
    __shared__ float lds_w[CC * KK];            // 27 conv weights

    const int b = blockIdx.x;
    const int t = threadIdx.x;                  // 0..255, wave32 -> 8 waves

    // Broadcast the tiny weight tensor through LDS (published by the barrier below).
    if (t < CC * KK) lds_w[t] = weight[t];

    // ---------------------------------------------------------------
    // Stage 1: TDM async DMA of x[b] (48 KB contiguous) global -> LDS.
    // Only wave 0 issues the tensor op (TDM ignores EXEC; descriptor is
    // wave-uniform -> lives in SGPRs). Tracked by TENSORcnt.
    // ---------------------------------------------------------------
    if (t < 32) {
        const uint64_t gaddr = (uint64_t)(uintptr_t)(x + (size_t)b * IMG_ELEMS);
        // Low 32 bits of the flat shared-aperture address == LDS byte offset.
        const uint32_t laddr = (uint32_t)(uintptr_t)(void*)lds_x;

        u32x4 g0;
        g0[0] = 1u;                                                  // count=1 (valid user D#)
        g0[1] = laddr;                                               // lds_addr
        g0[2] = (uint32_t)(gaddr & 0xFFFFFFFFu);                     // global_addr[31:0]
        g0[3] = (uint32_t)((gaddr >> 32) & 0x1FFFFFFu) | (2u << 30); // global_addr[56:32] | type=2

        i32x8 g1;
        g1[0] = (int)(2u << 16);           // data_size=2 (4 B), no multicast/pad/iterate
        g1[1] = (int)(12288u << 16);       // tensor_dim0[15:0]=12288 in bits [63:48]
        g1[2] = (int)(1u << 16);           // tensor_dim0 hi=0 ; tensor_dim1=1
        g1[3] = (int)(12288u << 16);       // tile_dim0 = 12288 elements (one row)
        g1[4] = 1;                         // tile_dim1 = 1, tile_dim2 = 0
        g1[5] = 12288;                     // tensor_dim0_stride (elements)
        g1[6] = 0;                         // stride hi / dim1_stride lo
        g1[7] = 0;                         // dim1_stride hi
        i32x4 gz4 = {0, 0, 0, 0};          // groups 2/3 unused (<=2D tensor)
        i32x8 gz8 = {0, 0, 0, 0, 0, 0, 0, 0};

        __builtin_amdgcn_tensor_load_to_lds(g0, g1, gz4, gz4, gz8, 0);
    }

    // ---------------------------------------------------------------
    // Overlap with DMA: every thread pulls its per-pixel offsets/mask
    // (fully coalesced: lane p reads element p of each 256-float plane).
    // ---------------------------------------------------------------
    float dyv[KK], dxv[KK], mkv[KK];
    {
        const float* op = offset + (size_t)b * (2 * KK * OUTHW * OUTHW) + t;
        const float* mp = mask   + (size_t)b * (KK * OUTHW * OUTHW)     + t;
#pragma unroll
        for (int k = 0; k < KK; ++k) {
            dyv[k] = op[(2 * k    ) * (OUTHW * OUTHW)];
            dxv[k] = op[(2 * k + 1) * (OUTHW * OUTHW)];
            mkv[k] = mp[ k          * (OUTHW * OUTHW)];
        }
    }

    // Wave 0 drains the tensor DMA, then the barrier publishes lds_x/lds_w.
    if (t < 32) __builtin_amdgcn_s_wait_tensorcnt(0);
    __syncthreads();

    // ---------------------------------------------------------------
    // Stage 2: bilinear resize 3x64x64 -> 3x18x18, entirely in LDS.
    // (align_corners=False: src = (o+0.5)*64/18 - 0.5, clipped)
    // ---------------------------------------------------------------
    const float scale = (float)IN_H / (float)RS;
    for (int idx = t; idx < CC * RS * RS; idx += 256) {
        int c  = idx / (RS * RS);
        int r  = idx - c * (RS * RS);
        int oy = r / RS;
        int ox = r - oy * RS;
        float sy = ((float)oy + 0.5f) * scale - 0.5f;
        float sx = ((float)ox + 0.5f) * scale - 0.5f;
        sy = fminf(fmaxf(sy, 0.0f), (float)(IN_H - 1));
        sx = fminf(fmaxf(sx, 0.0f), (float)(IN_W - 1));
        int y0 = (int)sy;                  // sy >= 0 -> trunc == floor
        int x0 = (int)sx;
        int y1 = min(y0 + 1, IN_H - 1);
        int x1 = min(x0 + 1, IN_W - 1);
        float wy = sy - (float)y0;
        float wx = sx - (float)x0;
        const float* img = lds_x + c * (IN_H * IN_W);
        float v00 = img[y0 * IN_W + x0], v01 = img[y0 * IN_W + x1];
        float v10 = img[y1 * IN_W + x0], v11 = img[y1 * IN_W + x1];
        float top = v00 + (v01 - v00) * wx;
        float bot = v10 + (v11 - v10) * wx;
        lds_r[idx] = top + (bot - top) * wy;
    }
    __syncthreads();

    // ---------------------------------------------------------------
    // Stage 3: modulated deformable 3x3 conv, one output pixel per thread.
    // 9 taps x 3 channels x 4 bilinear corners, branchless (validity folded
    // into the corner weights, indices clamped for the LDS gathers).
    // ---------------------------------------------------------------
    const int oy = t >> 4;
    const int ox = t & 15;
    float acc = 0.0f;
#pragma unroll
    for (int k = 0; k < KK; ++k) {
        const int ky = k / 3;
        const int kx = k - 3 * ky;
        float py = dyv[k] + (float)(oy + ky);
        float px = dxv[k] + (float)(ox + kx);
        float fy = floorf(py), fx = floorf(px);
        int y0 = (int)fy, x0 = (int)fx;
        int y1 = y0 + 1,  x1 = x0 + 1;
        float wy = py - fy, wx = px - fx;
        float vy0 = (y0 >= 0 && y0 < RS) ? 1.0f : 0.0f;
        float vy1 = (y1 >= 0 && y1 < RS) ? 1.0f : 0.0f;
        float vx0 = (x0 >= 0 && x0 < RS) ? 1.0f : 0.0f;
        float vx1 = (x1 >= 0 && x1 < RS) ? 1.0f : 0.0f;
        int yc0 = min(max(y0, 0), RS - 1), yc1 = min(max(y1, 0), RS - 1);
        int xc0 = min(max(x0, 0), RS - 1), xc1 = min(max(x1, 0), RS - 1);
        float w00 = (1.0f - wy) * (1.0f - wx) * vy0 * vx0;
        float w01 = (1.0f - wy) * wx          * vy0 * vx1;
        float w10 = wy          * (1.0f - wx) * vy1 * vx0;
        float w11 = wy          * wx          * vy1 * vx1;
        int a00 = yc0 * RS + xc0, a01 = yc0 * RS + xc1;
        int a10 = yc1 * RS + xc0, a11 = yc1 * RS + xc1;
        float mk = mkv[k];
#pragma unroll
        for (int c = 0; c < CC; ++c) {
            const float* r = lds_r + c * (RS * RS);
            float s = r[a00] * w00 + r[a01] * w01 + r[a10] * w10 + r[a11] * w11;
            acc += mk * lds_w[c * KK + k] * s;
        }
    }
    out[(size_t)b * (OUTHW * OUTHW) + t] = acc;
}

extern "C" void kernel_launch(void* const* d_in, const int* in_sizes, int n_in,
                              void* d_out, int out_size, void* d_ws, size_t ws_size,
                              hipStream_t stream) {
    (void)n_in; (void)out_size; (void)d_ws; (void)ws_size;
    const float* x      = (const float*)d_in[0];
    const float* offset = (const float*)d_in[1];
    const float* mask   = (const float*)d_in[2];
    const float* weight = (const float*)d_in[3];
    float* out = (float*)d_out;
    const int B = in_sizes[0] / IMG_ELEMS;   // 4096
    deform_resize_kernel<<<B, 256, 0, stream>>>(x, offset, mask, weight, out);
}